// ScaleGaussianLasryLions_33131377722052
// MI455X (gfx1250) — compile-verified
//
#include <hip/hip_runtime.h>
#include <cstdint>
#include <cstddef>

// ============================================================================
// ScaleGaussianLasryLions for MI455X (gfx1250, wave32)
//
// Per scale (w in {1,3,8,18}, K=2w+1), per channel c:
//   k[j]  = filt[j]*cm[c],  kc[j] = k[j]*cval[c],  filt symmetric
//   BLUR  = cm^2 * (filt (x) filt) * x          (channel-uniform conv -> WMMA)
//   DIL2  = maxpool2d(x + k),  ERO2 = minpool2d(x - k)
//   CLO   = minpool2d(DIL2 - kc), OPN = maxpool2d(ERO2 + kc)
//   out   = a0*CLO + a1*OPN + a1*BLUR
// Edge padding == per-pass clamped indexing (pad replicates whole rows/cols,
// and a 1-D pass of a replicated row equals the replicated result).
// ============================================================================

#define H_ 256
#define W_ 256
#define S_ 4
#define C_ 32
#define B_ 4

typedef float v2f __attribute__((ext_vector_type(2)));
typedef float v8f __attribute__((ext_vector_type(8)));

__device__ __forceinline__ float filt_at(int j, int wid, float inv) {
  float d = (float)(j - wid);
  return -d * d * inv;
}

// ---------------- CDNA5 feature paths (compile-guarded) ---------------------
#if defined(__gfx1250__) && __has_builtin(__builtin_amdgcn_global_load_async_to_lds_b128)
#define USE_ASYNC_LDS 1
#endif
#if defined(__gfx1250__) && __has_builtin(__builtin_amdgcn_wmma_f32_16x16x4_f32)
#define USE_WMMA 1
#endif

#ifdef USE_ASYNC_LDS
typedef int v4i __attribute__((vector_size(16)));
typedef __attribute__((address_space(1))) v4i as1_v4i;
typedef __attribute__((address_space(3))) v4i as3_v4i;
__device__ __forceinline__ void async_b128(const float* g, float* l) {
  __builtin_amdgcn_global_load_async_to_lds_b128((as1_v4i*)g, (as3_v4i*)l, 0,
                                                 0);
}
__device__ __forceinline__ void async_wait() {
  asm volatile("s_wait_asynccnt 0x0" ::: "memory");
}
#endif

// D = A(16x4) * B(4x16) + C(16x16), f32. Fragment layouts per CDNA5 ISA 7.12.2:
//  A: lane m=lane&15 holds K = (lane<16 ? {0,1} : {2,3}) in {a.x,a.y}
//  B: lane n=lane&15 holds K = (lane<16 ? {0,1} : {2,3}) in {b.x,b.y}
//  C/D: vgpr r holds (m = r + (lane<16?0:8), n = lane&15)
__device__ __forceinline__ v8f wmma16x16x4(v2f a, v2f b, v8f c, int lane) {
#ifdef USE_WMMA
  (void)lane;
  return __builtin_amdgcn_wmma_f32_16x16x4_f32(false, a, false, b, (short)0, c,
                                               false, false);
#else
  // Cross-lane emulation (host pass / missing builtin only).
  v8f d = c;
  int l15 = lane & 15;
#pragma unroll
  for (int r = 0; r < 8; ++r) {
    int m = r + ((lane >> 4) << 3);
    float sum = 0.f;
#pragma unroll
    for (int k = 0; k < 4; ++k) {
      int off = (k & 2) ? 16 : 0;
      float av = __shfl((k & 1) ? a.y : a.x, (m & 15) + off, 32);
      float bv = __shfl((k & 1) ? b.y : b.x, l15 + off, 32);
      sum += av * bv;
    }
    d[r] += sum;
  }
  return d;
#endif
}

// ============================================================================
// BLUR kernel: one wave -> one 16x16 pixel tile of one channel.
// H-pass: T(16 x IR) = band(filt) x X   via WMMA chunks of 4 input rows.
// W-pass: OUT = T x band(filt)          via WMMA chunks of 4 input cols.
// Writes a1*cm^2*OUT into d_out (initializes the output slice).
// ============================================================================
template <int WID>
__global__ __launch_bounds__(128) void k_blur(const float* __restrict__ x,
                                              const float* __restrict__ cm,
                                              const float* __restrict__ alpha,
                                              float* __restrict__ out,
                                              int scale, float inv) {
  constexpr int IR = 16 + 2 * WID;      // input extent (rows/cols) incl. halo
  constexpr int NT = (IR + 15) / 16;    // T column tiles
  constexpr int NR = (IR + 3) / 4;      // K-chunks of 4
  constexpr int TW = NT * 16;           // LDS T width
  __shared__ float Tlds[4][16 * TW];

  const int lane = threadIdx.x;
  const int wave = threadIdx.y;
  const int wb = blockIdx.x * 16;
  const int hb = blockIdx.y * 16;
  const int b = blockIdx.z >> 3;
  const int c = ((blockIdx.z & 7) << 2) + wave;
  float* T = &Tlds[wave][0];

  for (int u = lane; u < 16 * TW; u += 32) T[u] = 0.f;

  const int l15 = lane & 15;
  const int kk0 = (lane >> 4) << 1;  // 0 or 2

  // ---- H pass ----
  for (int ct = 0; ct < NT; ++ct) {
    int gc = wb - WID + ct * 16 + l15;
    gc = min(max(gc, 0), W_ - 1);
    v8f acc = {0.f, 0.f, 0.f, 0.f, 0.f, 0.f, 0.f, 0.f};
    for (int rc = 0; rc < NR; ++rc) {
      const int rr0 = rc * 4 + kk0;
      // A: coefficient band, M = l15 (output row h), K = rr0, rr0+1
      const int j0 = rr0 - l15, j1 = j0 + 1;
      v2f a;
      a.x = (j0 >= 0 && j0 <= 2 * WID) ? filt_at(j0, WID, inv) : 0.f;
      a.y = (j1 >= 0 && j1 <= 2 * WID) ? filt_at(j1, WID, inv) : 0.f;
      // B: input rows rr0, rr0+1 at column gc (clamped = edge pad)
      const int g0 = min(max(hb - WID + rr0, 0), H_ - 1);
      const int g1 = min(max(hb - WID + rr0 + 1, 0), H_ - 1);
      v2f bm;
      bm.x = x[(((size_t)b * H_ + g0) * W_ + gc) * (S_ * C_) +
               (size_t)scale * C_ + c];
      bm.y = x[(((size_t)b * H_ + g1) * W_ + gc) * (S_ * C_) +
               (size_t)scale * C_ + c];
      acc = wmma16x16x4(a, bm, acc, lane);
    }
#pragma unroll
    for (int r = 0; r < 8; ++r)
      T[(r + ((lane >> 4) << 3)) * TW + ct * 16 + l15] = acc[r];
  }
  // same-wave DS write->read: hardware keeps LDS ops in order (DScnt waits
  // inserted by the compiler); no workgroup barrier needed (private slice).

  // ---- W pass ----
  v8f acc = {0.f, 0.f, 0.f, 0.f, 0.f, 0.f, 0.f, 0.f};
  for (int rc = 0; rc < NR; ++rc) {
    const int ww0 = rc * 4 + kk0;
    // A: T rows, M = l15 (= h), K = ww0, ww0+1
    v2f a;
    a.x = T[l15 * TW + ww0];
    a.y = T[l15 * TW + ww0 + 1];
    // B: coefficient band, N = l15 (= output col w)
    const int j0 = ww0 - l15, j1 = j0 + 1;
    v2f bc;
    bc.x = (j0 >= 0 && j0 <= 2 * WID) ? filt_at(j0, WID, inv) : 0.f;
    bc.y = (j1 >= 0 && j1 <= 2 * WID) ? filt_at(j1, WID, inv) : 0.f;
    acc = wmma16x16x4(a, bc, acc, lane);
  }

  const float cmv = cm[c];
  const float sc = alpha[c * 3 + 1] * cmv * cmv;  // a1 * cm^2
#pragma unroll
  for (int r = 0; r < 8; ++r) {
    const int h = hb + r + ((lane >> 4) << 3);
    const int w = wb + l15;
    out[((((size_t)b * H_ + h) * W_ + w) * S_ + scale) * C_ + c] = sc * acc[r];
  }
}

// ============================================================================
// Morphology passes: block = one 1-D strip (64 outputs) x 32 channels.
// lane = channel -> every global access is a coalesced 128B pixel line.
// Strips staged in LDS via async global->LDS (halo by clamped addressing).
// Field layout: f[((bl*H + h)*W + w)*32 + c]
// ============================================================================
template <int WID>
__global__ __launch_bounds__(256) void k_hpass1(
    const float* __restrict__ x, const float* __restrict__ cm,
    float* __restrict__ fd, float* __restrict__ fe, int scale, float inv,
    int b0) {
  constexpr int K = 2 * WID + 1;
  constexpr int ROWS = 64;
  constexpr int LR = ROWS + 2 * WID;
  __shared__ float tile[LR * C_];
  const int w = blockIdx.x;
  const int h0 = blockIdx.y * ROWS;
  const int bl = blockIdx.z;
  const int b = b0 + bl;
  const int tid = threadIdx.y * 32 + threadIdx.x;

#ifdef USE_ASYNC_LDS
  for (int u = tid; u < LR * 8; u += 256) {
    const int r = u >> 3, sub = u & 7;
    const int gh = min(max(h0 + r - WID, 0), H_ - 1);
    const float* src = x + ((((size_t)b * H_ + gh) * W_ + w) * S_ + scale) * C_ +
                       sub * 4;
    async_b128(src, &tile[r * C_ + sub * 4]);
  }
  async_wait();
#else
  for (int u = tid; u < LR * C_; u += 256) {
    const int r = u >> 5, cc = u & 31;
    const int gh = min(max(h0 + r - WID, 0), H_ - 1);
    tile[u] = x[((((size_t)b * H_ + gh) * W_ + w) * S_ + scale) * C_ + cc];
  }
#endif
  __syncthreads();

  const int c = threadIdx.x;
  const float cmv = cm[c];
  for (int q = 0; q < ROWS / 8; ++q) {
    const int lr = threadIdx.y + 8 * q;
    float d = -3.402823466e38f, e = 3.402823466e38f;
#pragma unroll
    for (int j = 0; j < K; ++j) {
      const float kj = filt_at(j, WID, inv) * cmv;
      const float v = tile[(lr + j) * C_ + c];
      d = fmaxf(d, v + kj);
      e = fminf(e, v - kj);
    }
    const size_t o = (((size_t)bl * H_ + (h0 + lr)) * W_ + w) * C_ + c;
    fd[o] = d;
    fe[o] = e;
  }
}

template <int WID>
__global__ __launch_bounds__(256) void k_wpass1(
    const float* __restrict__ fd1, const float* __restrict__ fe1,
    const float* __restrict__ cm, float* __restrict__ fdil,
    float* __restrict__ fero, float inv) {
  constexpr int K = 2 * WID + 1;
  constexpr int COLS = 64;
  constexpr int LC = COLS + 2 * WID;
  __shared__ float td[LC * C_];
  __shared__ float te[LC * C_];
  const int h = blockIdx.x;
  const int w0 = blockIdx.y * COLS;
  const int bl = blockIdx.z;
  const int tid = threadIdx.y * 32 + threadIdx.x;
  const size_t rowbase = ((size_t)bl * H_ + h) * W_;

#ifdef USE_ASYNC_LDS
  for (int u = tid; u < LC * 8; u += 256) {
    const int r = u >> 3, sub = u & 7;
    const int gw = min(max(w0 + r - WID, 0), W_ - 1);
    const size_t src = (rowbase + gw) * C_ + sub * 4;
    async_b128(fd1 + src, &td[r * C_ + sub * 4]);
    async_b128(fe1 + src, &te[r * C_ + sub * 4]);
  }
  async_wait();
#else
  for (int u = tid; u < LC * C_; u += 256) {
    const int r = u >> 5, cc = u & 31;
    const int gw = min(max(w0 + r - WID, 0), W_ - 1);
    td[u] = fd1[(rowbase + gw) * C_ + cc];
    te[u] = fe1[(rowbase + gw) * C_ + cc];
  }
#endif
  __syncthreads();

  const int c = threadIdx.x;
  const float cmv = cm[c];
  for (int q = 0; q < COLS / 8; ++q) {
    const int lw = threadIdx.y + 8 * q;
    float dmx = -3.402823466e38f, emn = 3.402823466e38f;
#pragma unroll
    for (int j = 0; j < K; ++j) {
      const float kj = filt_at(j, WID, inv) * cmv;
      dmx = fmaxf(dmx, td[(lw + j) * C_ + c] + kj);
      emn = fminf(emn, te[(lw + j) * C_ + c] - kj);
    }
    const size_t o = (rowbase + (w0 + lw)) * C_ + c;
    fdil[o] = dmx;
    fero[o] = emn;
  }
}

template <int WID>
__global__ __launch_bounds__(256) void k_hpass2(
    const float* __restrict__ fdil, const float* __restrict__ fero,
    const float* __restrict__ cm, const float* __restrict__ cv,
    float* __restrict__ fce, float* __restrict__ fod, float inv) {
  constexpr int K = 2 * WID + 1;
  constexpr int ROWS = 64;
  constexpr int LR = ROWS + 2 * WID;
  __shared__ float td[LR * C_];
  __shared__ float te[LR * C_];
  const int w = blockIdx.x;
  const int h0 = blockIdx.y * ROWS;
  const int bl = blockIdx.z;
  const int tid = threadIdx.y * 32 + threadIdx.x;

#ifdef USE_ASYNC_LDS
  for (int u = tid; u < LR * 8; u += 256) {
    const int r = u >> 3, sub = u & 7;
    const int gh = min(max(h0 + r - WID, 0), H_ - 1);
    const size_t src = (((size_t)bl * H_ + gh) * W_ + w) * C_ + sub * 4;
    async_b128(fdil + src, &td[r * C_ + sub * 4]);
    async_b128(fero + src, &te[r * C_ + sub * 4]);
  }
  async_wait();
#else
  for (int u = tid; u < LR * C_; u += 256) {
    const int r = u >> 5, cc = u & 31;
    const int gh = min(max(h0 + r - WID, 0), H_ - 1);
    td[u] = fdil[(((size_t)bl * H_ + gh) * W_ + w) * C_ + cc];
    te[u] = fero[(((size_t)bl * H_ + gh) * W_ + w) * C_ + cc];
  }
#endif
  __syncthreads();

  const int c = threadIdx.x;
  const float kc_s = cm[c] * cv[c];
  for (int q = 0; q < ROWS / 8; ++q) {
    const int lr = threadIdx.y + 8 * q;
    float ce = 3.402823466e38f, od = -3.402823466e38f;
#pragma unroll
    for (int j = 0; j < K; ++j) {
      const float kj = filt_at(j, WID, inv) * kc_s;
      ce = fminf(ce, td[(lr + j) * C_ + c] - kj);  // erode(DIL2, kc) along H
      od = fmaxf(od, te[(lr + j) * C_ + c] + kj);  // dilate(ERO2, kc) along H
    }
    const size_t o = (((size_t)bl * H_ + (h0 + lr)) * W_ + w) * C_ + c;
    fce[o] = ce;
    fod[o] = od;
  }
}

template <int WID>
__global__ __launch_bounds__(256) void k_wpass2(
    const float* __restrict__ fce, const float* __restrict__ fod,
    const float* __restrict__ cm, const float* __restrict__ cv,
    const float* __restrict__ alpha, float* __restrict__ out, int scale,
    float inv, int b0) {
  constexpr int K = 2 * WID + 1;
  constexpr int COLS = 64;
  constexpr int LC = COLS + 2 * WID;
  __shared__ float td[LC * C_];
  __shared__ float te[LC * C_];
  const int h = blockIdx.x;
  const int w0 = blockIdx.y * COLS;
  const int bl = blockIdx.z;
  const int b = b0 + bl;
  const int tid = threadIdx.y * 32 + threadIdx.x;
  const size_t rowbase = ((size_t)bl * H_ + h) * W_;

#ifdef USE_ASYNC_LDS
  for (int u = tid; u < LC * 8; u += 256) {
    const int r = u >> 3, sub = u & 7;
    const int gw = min(max(w0 + r - WID, 0), W_ - 1);
    const size_t src = (rowbase + gw) * C_ + sub * 4;
    async_b128(fce + src, &td[r * C_ + sub * 4]);
    async_b128(fod + src, &te[r * C_ + sub * 4]);
  }
  async_wait();
#else
  for (int u = tid; u < LC * C_; u += 256) {
    const int r = u >> 5, cc = u & 31;
    const int gw = min(max(w0 + r - WID, 0), W_ - 1);
    td[u] = fce[(rowbase + gw) * C_ + cc];
    te[u] = fod[(rowbase + gw) * C_ + cc];
  }
#endif
  __syncthreads();

  const int c = threadIdx.x;
  const float kc_s = cm[c] * cv[c];
  const float a0 = alpha[c * 3 + 0];
  const float a1 = alpha[c * 3 + 1];
  for (int q = 0; q < COLS / 8; ++q) {
    const int lw = threadIdx.y + 8 * q;
    float clo = 3.402823466e38f, opn = -3.402823466e38f;
#pragma unroll
    for (int j = 0; j < K; ++j) {
      const float kj = filt_at(j, WID, inv) * kc_s;
      clo = fminf(clo, td[(lw + j) * C_ + c] - kj);
      opn = fmaxf(opn, te[(lw + j) * C_ + c] + kj);
    }
    const size_t o =
        ((((size_t)b * H_ + h) * W_ + (w0 + lw)) * S_ + scale) * C_ + c;
    out[o] = a0 * clo + a1 * opn + out[o];  // blur term already in out
  }
}

// ============================================================================
static inline int imin_(int a, int b) { return a < b ? a : b; }

extern "C" void kernel_launch(void* const* d_in, const int* in_sizes, int n_in,
                              void* d_out, int out_size, void* d_ws,
                              size_t ws_size, hipStream_t stream) {
  (void)in_sizes; (void)n_in; (void)out_size;
  const float* x  = (const float*)d_in[0];
  const float* cm = (const float*)d_in[1];  // coef_morpho (1,C)
  // d_in[2] = coef_gaussian: unused by the reference math
  const float* cv = (const float*)d_in[3];  // c (1,1,C)
  const float* al = (const float*)d_in[4];  // alpha (1,1,1,C,3)
  float* out = (float*)d_out;

  const size_t FIELD_B = (size_t)H_ * W_ * C_;  // floats per field per batch
  int CB = (int)(ws_size / (4ull * FIELD_B * sizeof(float)));
  if (CB < 1) CB = 1;
  if (CB > B_) CB = B_;
  float* f0 = (float*)d_ws;
  float* f1 = f0 + (size_t)CB * FIELD_B;
  float* f2 = f1 + (size_t)CB * FIELD_B;
  float* f3 = f2 + (size_t)CB * FIELD_B;

  const dim3 blk(32, 8);

#define RUN_SCALE(WID, SC, INV)                                                \
  do {                                                                         \
    dim3 bb(32, 4), gb(W_ / 16, H_ / 16, B_ * (C_ / 4));                       \
    k_blur<WID><<<gb, bb, 0, stream>>>(x, cm, al, out, SC, INV);               \
    for (int b0 = 0; b0 < B_; b0 += CB) {                                      \
      int nb = imin_(CB, B_ - b0);                                             \
      dim3 gh(W_, H_ / 64, nb), gw(H_, W_ / 64, nb);                           \
      k_hpass1<WID><<<gh, blk, 0, stream>>>(x, cm, f0, f1, SC, INV, b0);       \
      k_wpass1<WID><<<gw, blk, 0, stream>>>(f0, f1, cm, f2, f3, INV);          \
      k_hpass2<WID><<<gh, blk, 0, stream>>>(f2, f3, cm, cv, f0, f1, INV);      \
      k_wpass2<WID><<<gw, blk, 0, stream>>>(f0, f1, cm, cv, al, out, SC, INV,  \
                                            b0);                               \
    }                                                                          \
  } while (0)

  // SCALES = {0.25, 0.75, 1.75, 3.75}; inv = 1/scale^2
  RUN_SCALE(1, 0, 16.0f);
  RUN_SCALE(3, 1, 1.77777778f);
  RUN_SCALE(8, 2, 0.32653061f);
  RUN_SCALE(18, 3, 0.07111111f);

#undef RUN_SCALE
}